// SumLayer_7962869366905
// MI455X (gfx1250) — compile-verified
//
#include <hip/hip_runtime.h>
#include <math.h>

typedef __attribute__((ext_vector_type(2))) float v2f;
typedef __attribute__((ext_vector_type(8))) float v8f;

#define R_DIM 1024
#define I_DIM 16
#define O_DIM 16
#define B_DIM 512
#define TILES_PER_WAVE 4   // 8 waves * 4 tiles = 32 b-tiles = all of B per block

// out[b,r,o] = mx[b] + log( sum_i exp(x[b,r,i]-mx) * exp(w[r,o,i]-mw[o]) ) - log(ssum[o])
// where ssum[o] = sum_i exp(w[r,o,i]-mw[o])  (softmax normalizer, kept out of the GEMM)
__global__ __launch_bounds__(256) void sumlayer_wmma_kernel(
    const float* __restrict__ x,       // (B=512, R=1024, I=16)
    const float* __restrict__ weight,  // (R=1024, O=16, I=16)
    float* __restrict__ out)           // (B=512, R=1024, O=16)
{
    const int r    = blockIdx.x;
    const int wave = threadIdx.x >> 5;
    const int lane = threadIdx.x & 31;
    const int lo   = lane & 15;   // M (A rows) / N (B cols) / out column o
    const int hi   = lane >> 4;   // half-wave selector (K sub-pair)

    // ---------------- B fragment: e_w[o][i] = exp(w[r][o][i] - mw[o]) -----
    // Lane holds B[k][n] = e_w[o=lo][i=k], k = Kbase + 2*hi + j, Kbase in {0,4,8,12}.
    const float* wrow = weight + (size_t)r * (O_DIM * I_DIM) + (size_t)lo * I_DIM;
    float wv[8];
#pragma unroll
    for (int kb = 0; kb < 4; ++kb) {
        v2f t = *(const v2f*)(wrow + kb * 4 + 2 * hi);
        wv[2 * kb + 0] = t.x;
        wv[2 * kb + 1] = t.y;
    }
    float mw = wv[0];
#pragma unroll
    for (int j = 1; j < 8; ++j) mw = fmaxf(mw, wv[j]);
    mw = fmaxf(mw, __shfl_xor(mw, 16, 32));   // full row max over I=16
    float ssum = 0.0f;
#pragma unroll
    for (int j = 0; j < 8; ++j) { wv[j] = __expf(wv[j] - mw); ssum += wv[j]; }
    ssum += __shfl_xor(ssum, 16, 32);         // full row sum over I=16
    const float lssum = __logf(ssum);         // per-lane: normalizer for o = lo
    v2f bfrag[4];
#pragma unroll
    for (int kb = 0; kb < 4; ++kb) {
        bfrag[kb].x = wv[2 * kb + 0];
        bfrag[kb].y = wv[2 * kb + 1];
    }

    // ---------------- loop over batch tiles of 16 rows --------------------
#pragma unroll
    for (int t = 0; t < TILES_PER_WAVE; ++t) {
        const int btile = wave * TILES_PER_WAVE + t;  // 0..31
        const int b0    = btile * 16;

        // A fragment: lane holds A[M=lo][k] = exp(x[b0+lo][r][k] - mx), same k set.
        const float* xrow = x + ((size_t)(b0 + lo) * R_DIM + r) * I_DIM;
        float xv[8];
#pragma unroll
        for (int kb = 0; kb < 4; ++kb) {
            v2f t2 = *(const v2f*)(xrow + kb * 4 + 2 * hi);
            xv[2 * kb + 0] = t2.x;
            xv[2 * kb + 1] = t2.y;
        }
        float mx = xv[0];
#pragma unroll
        for (int j = 1; j < 8; ++j) mx = fmaxf(mx, xv[j]);
        mx = fmaxf(mx, __shfl_xor(mx, 16, 32));   // row max over I=16

        v2f afrag[4];
#pragma unroll
        for (int kb = 0; kb < 4; ++kb) {
            afrag[kb].x = __expf(xv[2 * kb + 0] - mx);
            afrag[kb].y = __expf(xv[2 * kb + 1] - mx);
        }

        // S = A * B accumulated over K=16 as 4 chained K=4 WMMAs.
        v8f c = {0.f, 0.f, 0.f, 0.f, 0.f, 0.f, 0.f, 0.f};
#pragma unroll
        for (int kb = 0; kb < 4; ++kb) {
            c = __builtin_amdgcn_wmma_f32_16x16x4_f32(
                    /*neg_a=*/false, afrag[kb],
                    /*neg_b=*/false, bfrag[kb],
                    /*c_mod=*/(short)0, c,
                    /*reuse_a=*/false, /*reuse_b=*/false);
        }

        // D layout: lane holds C[brow = v + 8*hi][o = lo], v = 0..7.
        // out[b0+brow, r, o] = mx[brow] + log(S) - log(ssum[o])
#pragma unroll
        for (int v = 0; v < 8; ++v) {
            const int brow = v + 8 * hi;
            const float m_b = __shfl(mx, brow, 32);  // lane 'brow' holds that row's max
            const float val = m_b + __logf(c[v]) - lssum;
            out[((size_t)(b0 + brow) * R_DIM + r) * O_DIM + lo] = val;
        }
    }
}

extern "C" void kernel_launch(void* const* d_in, const int* in_sizes, int n_in,
                              void* d_out, int out_size, void* d_ws, size_t ws_size,
                              hipStream_t stream) {
    const float* x = (const float*)d_in[0];       // (512, 1024, 16)
    const float* w = (const float*)d_in[1];       // (1024, 16, 16)
    float* out     = (float*)d_out;               // (512, 1024, 16)
    (void)in_sizes; (void)n_in; (void)out_size; (void)d_ws; (void)ws_size;

    dim3 grid(R_DIM, 1, 1);
    dim3 block(256, 1, 1);
    sumlayer_wmma_kernel<<<grid, block, 0, stream>>>(x, w, out);
}